// Clustering_43044162240699
// MI455X (gfx1250) — compile-verified
//
#include <hip/hip_runtime.h>
#include <math.h>

typedef __attribute__((ext_vector_type(2))) float v2f;
typedef __attribute__((ext_vector_type(8))) float v8f;

#define R_ 1000
#define C_ 2048
#define CH_ 32
#define NCHUNK 20
#define CHUNK 50   // NCHUNK*CHUNK == R_

// d_out layout (floats): [0,32000)=y ; 32000=max_cmd ; [32001,32033)=max_nnz_ch ;
//                        [32033,32065)=num_col_ch ; [32065,32097)=row_ch
#define O_MAXCMD 32000
#define O_NNZ    32001
#define O_NUMCOL 32033
#define O_ROWCH  32065

// ---------------------------------------------------------------------------
// K1: per-row softmax of Y -> out[0..31999]; per-row sum of D -> ws_rowsum[r]
// One wave (32 lanes) per row; lane == channel.
// ---------------------------------------------------------------------------
__global__ void k_softmax_rowsum(const float* __restrict__ Y,
                                 const float* __restrict__ D,
                                 float* __restrict__ out,
                                 float* __restrict__ rowsumD) {
    const int lane = threadIdx.x & 31;
    const int r = blockIdx.x * 8 + (threadIdx.x >> 5);   // 125 blocks * 8 waves = 1000

    // softmax over 32 channels
    float v = Y[r * CH_ + lane];
    float m = v;
    #pragma unroll
    for (int off = 16; off > 0; off >>= 1) m = fmaxf(m, __shfl_xor(m, off, 32));
    float e = expf(v - m);
    float s = e;
    #pragma unroll
    for (int off = 16; off > 0; off >>= 1) s += __shfl_xor(s, off, 32);
    out[r * CH_ + lane] = e / s;

    // row-sum of D (2048 cols, 64 strided passes, coalesced)
    float rs = 0.0f;
    #pragma unroll 4
    for (int k = 0; k < C_ / 32; ++k) rs += D[(size_t)r * C_ + (k << 5) + lane];
    #pragma unroll
    for (int off = 16; off > 0; off >>= 1) rs += __shfl_xor(rs, off, 32);
    if (lane == 0) rowsumD[r] = rs;
}

// ---------------------------------------------------------------------------
// K2: row_ch = y^T @ ones, nnz_ch = (y .* rowsumD)^T @ ones via WMMA f32 16x16x4.
// B = all-ones so any B lane layout gives the correct result; A/C layouts per ISA.
// Single wave (32 threads), EXEC all ones as WMMA requires.
// ---------------------------------------------------------------------------
__global__ void k_wmma_row_nnz(const float* __restrict__ y,
                               const float* __restrict__ rowsumD,
                               float* __restrict__ out) {
    const int lane = threadIdx.x;      // 0..31
    const int half = lane >> 4;        // A layout: lanes 16-31 hold K=2,3
    const int chA  = lane & 15;        // A layout: M = lane % 16

    v8f cr0 = {}, cn0 = {}, cr1 = {}, cn1 = {};
    const v2f ones = {1.0f, 1.0f};

    for (int it = 0; it < R_ / 4; ++it) {
        const int kk = it * 4 + 2 * half;            // K index for VGPR0; VGPR1 = kk+1
        const float rs0 = rowsumD[kk];
        const float rs1 = rowsumD[kk + 1];

        // channel block 0..15
        v2f a0 = { y[kk * CH_ + chA], y[(kk + 1) * CH_ + chA] };
        v2f a1 = { a0.x * rs0, a0.y * rs1 };
        // channel block 16..31
        v2f b0 = { y[kk * CH_ + 16 + chA], y[(kk + 1) * CH_ + 16 + chA] };
        v2f b1 = { b0.x * rs0, b0.y * rs1 };

        cr0 = __builtin_amdgcn_wmma_f32_16x16x4_f32(false, a0, false, ones, (short)0, cr0, false, false);
        cn0 = __builtin_amdgcn_wmma_f32_16x16x4_f32(false, a1, false, ones, (short)0, cn0, false, false);
        cr1 = __builtin_amdgcn_wmma_f32_16x16x4_f32(false, b0, false, ones, (short)0, cr1, false, false);
        cn1 = __builtin_amdgcn_wmma_f32_16x16x4_f32(false, b1, false, ones, (short)0, cn1, false, false);
    }

    // C layout: VGPR v, lanes 0-15 -> M=v ; lanes 16-31 -> M=v+8. All N columns equal.
    if (lane == 0) {
        #pragma unroll
        for (int v = 0; v < 8; ++v) {
            out[O_ROWCH + v]      = cr0[v];
            out[O_NNZ   + v]      = cn0[v];
            out[O_ROWCH + 16 + v] = cr1[v];
            out[O_NNZ   + 16 + v] = cn1[v];
        }
    }
    if (lane == 16) {
        #pragma unroll
        for (int v = 0; v < 8; ++v) {
            out[O_ROWCH + 8 + v]  = cr0[v];
            out[O_NNZ   + 8 + v]  = cn0[v];
            out[O_ROWCH + 24 + v] = cr1[v];
            out[O_NNZ   + 24 + v] = cn1[v];
        }
    }
}

// ---------------------------------------------------------------------------
// K3 (heavy): per-(c,ch) partial products over an r-chunk, accumulated in log
// space via atomicAdd into col_log[ch][c].
// lane = column (coalesced D loads); 32 channel accumulators as 16 packed f32
// pairs -> v_pk_fma_f32 / v_pk_mul_f32; y row fetched wave-uniform (scalar).
// ---------------------------------------------------------------------------
__global__ void k_colprod(const float* __restrict__ D,
                          const float* __restrict__ y,
                          float* __restrict__ col_log) {
    const int gtid  = blockIdx.x * blockDim.x + threadIdx.x;
    const int wave  = gtid >> 5;              // 1280 waves
    const int lane  = threadIdx.x & 31;
    const int group = wave & 63;              // 64 column groups of 32
    const int chunk = wave >> 6;              // 20 r-chunks of 50
    const int c  = (group << 5) + lane;
    const int r0 = chunk * CHUNK;

    v2f acc[16];
    #pragma unroll
    for (int i = 0; i < 16; ++i) acc[i] = (v2f){1.0f, 1.0f};

    for (int rr = 0; rr < CHUNK; ++rr) {
        const int r = r0 + rr;
        const float d = D[(size_t)r * C_ + c];
        if (r + 4 < R_) __builtin_prefetch(&D[(size_t)(r + 4) * C_ + c], 0, 1);
        const v2f* __restrict__ yr = (const v2f*)(y + r * CH_);  // wave-uniform -> s_load
        const v2f dd = {d, d};
        #pragma unroll
        for (int i = 0; i < 16; ++i) {
            v2f t = (v2f){1.0f, 1.0f} - yr[i] * dd;   // pk_fma
            acc[i] = acc[i] * t;                       // pk_mul
        }
    }

    #pragma unroll
    for (int i = 0; i < 16; ++i) {
        atomicAdd(&col_log[(2 * i)     * C_ + c], logf(acc[i].x));
        atomicAdd(&col_log[(2 * i + 1) * C_ + c], logf(acc[i].y));
    }
}

// ---------------------------------------------------------------------------
// K4: num_col_ch[ch] = sum_c (1 - exp(col_log[ch][c])). One block per channel.
// ---------------------------------------------------------------------------
__global__ void k_numcol(const float* __restrict__ col_log, float* __restrict__ out) {
    __shared__ float sdata[256];
    const int ch = blockIdx.x;
    const int t  = threadIdx.x;
    float s = 0.0f;
    for (int j = t; j < C_; j += 256) s += 1.0f - expf(col_log[ch * C_ + j]);
    sdata[t] = s;
    __syncthreads();
    for (int off = 128; off > 0; off >>= 1) {
        if (t < off) sdata[t] += sdata[t + off];
        __syncthreads();
    }
    if (t == 0) out[O_NUMCOL + ch] = sdata[0];
}

// ---------------------------------------------------------------------------
// K5: max_cmd = max_ch(nnz + num_col + row). One wave.
// ---------------------------------------------------------------------------
__global__ void k_final(float* __restrict__ out) {
    const int lane = threadIdx.x;
    float cmd = out[O_NNZ + lane] + out[O_NUMCOL + lane] + out[O_ROWCH + lane];
    #pragma unroll
    for (int off = 16; off > 0; off >>= 1) cmd = fmaxf(cmd, __shfl_xor(cmd, off, 32));
    if (lane == 0) out[O_MAXCMD] = cmd;
}

extern "C" void kernel_launch(void* const* d_in, const int* in_sizes, int n_in,
                              void* d_out, int out_size, void* d_ws, size_t ws_size,
                              hipStream_t stream) {
    const float* Y = (const float*)d_in[0];   // (1000, 32)
    const float* D = (const float*)d_in[1];   // (1000, 2048)
    float* out = (float*)d_out;

    float* col_log = (float*)d_ws;            // CH_*C_ = 65536 floats
    float* rowsumD = col_log + CH_ * C_;      // 1000 floats

    hipMemsetAsync(col_log, 0, (size_t)CH_ * C_ * sizeof(float), stream);

    k_softmax_rowsum<<<R_ / 8, 256, 0, stream>>>(Y, D, out, rowsumD);
    k_wmma_row_nnz<<<1, 32, 0, stream>>>(out, rowsumD, out);
    k_colprod<<<(64 * NCHUNK) / 8, 256, 0, stream>>>(D, out, col_log);
    k_numcol<<<CH_, 256, 0, stream>>>(col_log, out);
    k_final<<<1, 32, 0, stream>>>(out);
}